// SPGformer_54073638257177
// MI455X (gfx1250) — compile-verified
//
#include <hip/hip_runtime.h>
#include <math.h>

#define N_PIX 65536
#define S_SP  1024
#define HIDE_ 128
#define OUT_  64
#define CLS_  16
#define E_A_  16384
#define E_IMP_ 6144
#define E_PX_ 1048576
#define EPS_  1e-5f

typedef float v2f __attribute__((ext_vector_type(2)));
typedef float v8f __attribute__((ext_vector_type(8)));

__device__ __forceinline__ v8f wmma4(v2f a, v2f b, v8f c) {
  // D = A(16x4 f32) * B(4x16 f32) + C(16x16 f32), wave32
  return __builtin_amdgcn_wmma_f32_16x16x4_f32(false, a, false, b, (short)0, c,
                                               false, false);
}

// monotonic float<->uint for atomicMax on floats
__device__ __forceinline__ unsigned fmono(float f) {
  unsigned u = __float_as_uint(f);
  return (u >> 31) ? ~u : (u | 0x80000000u);
}
__device__ __forceinline__ float funmono(unsigned m) {
  return (m >> 31) ? __uint_as_float(m ^ 0x80000000u) : __uint_as_float(~m);
}

// ---------------------------------------------------------------------------
// Generic fp32 WMMA GEMM:  out[M,N] = A[M,K] @ W[K,N] (+ bias) (+ BN)
// grid.x = M/16, blockDim = 32 * (N/16) (one wave per 16-col tile)
// mode 0: (+bias); mode 1: (+bias) then BatchNorm (eval).  out2: optional copy.
// ---------------------------------------------------------------------------
__global__ void gemm_wmma(const float* __restrict__ A, const float* __restrict__ W,
                          const float* __restrict__ bias, float* __restrict__ out,
                          int K, int N, int mode,
                          const float* __restrict__ bng, const float* __restrict__ bnb,
                          const float* __restrict__ bnm, const float* __restrict__ bnv,
                          float* __restrict__ out2) {
  const int lane = threadIdx.x & 31;
  const int wv   = threadIdx.x >> 5;
  const int l15  = lane & 15;
  const int hi   = lane >> 4;          // 0: K,K+1   1: K+2,K+3
  const int m0   = blockIdx.x << 4;
  const int n0   = wv << 4;

  const float* ap = A + (size_t)(m0 + l15) * K + 2 * hi;
  const float* bp = W + (size_t)(2 * hi) * N + n0 + l15;

  v8f c = {};
  #pragma unroll 8
  for (int k = 0; k < K; k += 4) {
    v2f a = *(const v2f*)(ap + k);                 // A[m, k+2hi .. +1] contiguous
    v2f b;
    b.x = bp[(size_t)k * N];                       // W[k+2hi,   col]
    b.y = bp[(size_t)(k + 1) * N];                 // W[k+2hi+1, col]
    c = wmma4(a, b, c);
  }

  const int col = n0 + l15;
  float bs = bias ? bias[col] : 0.f;
  float sc = 1.f, sh = 0.f;
  if (mode == 1) {
    sc = bng[col] * rsqrtf(bnv[col] + EPS_);
    sh = bnb[col] - bnm[col] * sc;
  }
  #pragma unroll
  for (int i = 0; i < 8; ++i) {
    int row = m0 + (hi << 3) + i;
    float y = c[i] + bs;
    if (mode == 1) y = y * sc + sh;
    out[(size_t)row * N + col] = y;
    if (out2) out2[(size_t)row * N + col] = y;
  }
}

// ---------------------------------------------------------------------------
// QtX[s,h] += sum_k Q[k,s] * x0[k,h]   (split-K, atomic reduce)
// grid = (S/16, nchunks), blockDim = 256 (8 waves = 8 h-tiles)
// ---------------------------------------------------------------------------
__global__ void qtx_wmma(const float* __restrict__ Q, const float* __restrict__ X,
                         float* __restrict__ out, int kchunk) {
  const int lane = threadIdx.x & 31;
  const int wv   = threadIdx.x >> 5;
  const int l15  = lane & 15;
  const int hi   = lane >> 4;
  const int s0   = blockIdx.x << 4;
  const int h0   = wv << 4;
  const int kb   = blockIdx.y * kchunk;

  const float* qp = Q + (size_t)(kb + 2 * hi) * S_SP + s0 + l15;
  const float* xp = X + (size_t)(kb + 2 * hi) * HIDE_ + h0 + l15;

  v8f c = {};
  #pragma unroll 8
  for (int k = 0; k < kchunk; k += 4) {
    v2f a; a.x = qp[(size_t)k * S_SP]; a.y = qp[(size_t)(k + 1) * S_SP];
    v2f b; b.x = xp[(size_t)k * HIDE_]; b.y = xp[(size_t)(k + 1) * HIDE_];
    c = wmma4(a, b, c);
  }
  #pragma unroll
  for (int i = 0; i < 8; ++i) {
    int row = s0 + (hi << 3) + i;
    atomicAdd(&out[(size_t)row * HIDE_ + h0 + l15], c[i]);
  }
}

// column sums of Q  (grid = (S/256, 64), block = 256)
__global__ void colsum_q(const float* __restrict__ Q, float* __restrict__ Qs) {
  int s  = blockIdx.x * 256 + threadIdx.x;
  int r0 = blockIdx.y * (N_PIX / 64);
  float acc = 0.f;
  for (int r = r0; r < r0 + N_PIX / 64; ++r) acc += Q[(size_t)r * S_SP + s];
  atomicAdd(&Qs[s], acc);
}

// hp = QtX / Qs[row]
__global__ void scale_hp(const float* __restrict__ QtX, const float* __restrict__ Qs,
                         float* __restrict__ hp) {
  int i = blockIdx.x * 256 + threadIdx.x;
  if (i < S_SP * HIDE_) hp[i] = QtX[i] / Qs[i >> 7];
}

// out[dst] += val * feat[src]   (128-wide, one wave per edge, float4/lane)
__global__ void edge_scatter128(const float* __restrict__ feat,
                                const float* __restrict__ vals,
                                const int* __restrict__ src, const int* __restrict__ dst,
                                int E, float* __restrict__ out) {
  int t = blockIdx.x * 256 + threadIdx.x;
  int e = t >> 5, lane = t & 31;
  if (e >= E) return;
  float v = vals[e];
  int s = src[e], d = dst[e];
  float4 h = *(const float4*)(feat + (size_t)s * 128 + lane * 4);
  float* o = out + (size_t)d * 128 + lane * 4;
  atomicAdd(o + 0, v * h.x);
  atomicAdd(o + 1, v * h.y);
  atomicAdd(o + 2, v * h.z);
  atomicAdd(o + 3, v * h.w);
}

// y = lrelu(BN(x)), 128 channels
__global__ void bn_lrelu128(const float* __restrict__ in, float* __restrict__ out,
                            const float* __restrict__ g, const float* __restrict__ b,
                            const float* __restrict__ m, const float* __restrict__ v,
                            int n) {
  int i = blockIdx.x * 256 + threadIdx.x;
  if (i >= n) return;
  int c = i & 127;
  float y = (in[i] - m[c]) * (g[c] * rsqrtf(v[c] + EPS_)) + b[c];
  out[i] = y >= 0.f ? y : 0.01f * y;
}

// in-place layernorm over 64 cols, one wave per row
__global__ void ln64(float* __restrict__ X, int rows) {
  int w = (blockIdx.x * blockDim.x + threadIdx.x) >> 5;
  int lane = threadIdx.x & 31;
  if (w >= rows) return;
  float2 v = *(const float2*)(X + (size_t)w * 64 + lane * 2);
  float s = v.x + v.y;
  for (int o = 1; o < 32; o <<= 1) s += __shfl_xor(s, o);
  float mu = s * (1.f / 64.f);
  float dx = v.x - mu, dy = v.y - mu;
  float q = dx * dx + dy * dy;
  for (int o = 1; o < 32; o <<= 1) q += __shfl_xor(q, o);
  float rstd = rsqrtf(q * (1.f / 64.f) + EPS_);
  float2 r; r.x = dx * rstd; r.y = dy * rstd;
  *(float2*)(X + (size_t)w * 64 + lane * 2) = r;
}

// per-edge scaled dot(64) + segment max; 16 lanes/edge
__global__ void edge_scores(const float* __restrict__ Qf, const int* __restrict__ src,
                            const int* __restrict__ dst, float* __restrict__ sc,
                            unsigned* __restrict__ maxu, int E) {
  int t = blockIdx.x * 256 + threadIdx.x;
  int e = t >> 4, sub = t & 15;
  if (e >= E) return;
  int s = src[e], d = dst[e];
  float4 qa = *(const float4*)(Qf + (size_t)s * 64 + sub * 4);
  float4 qb = *(const float4*)(Qf + (size_t)d * 64 + sub * 4);
  float p = qa.x * qb.x + qa.y * qb.y + qa.z * qb.z + qa.w * qb.w;
  p += __shfl_xor(p, 1); p += __shfl_xor(p, 2);
  p += __shfl_xor(p, 4); p += __shfl_xor(p, 8);
  if (sub == 0) {
    float v = p * (1.0f / 64.0f);
    sc[e] = v;
    atomicMax(&maxu[s], fmono(v));
  }
}

// sc[e] = exp(sc[e]-max[src]); sum[src] += sc[e]
__global__ void edge_expsum(float* __restrict__ sc, const int* __restrict__ src,
                            const unsigned* __restrict__ maxu, float* __restrict__ sum,
                            int E) {
  int e = blockIdx.x * 256 + threadIdx.x;
  if (e >= E) return;
  int s = src[e];
  float ex = expf(sc[e] - funmono(maxu[s]));
  sc[e] = ex;
  atomicAdd(&sum[s], ex);
}

// out[src, colbase..+63] += (sc[e]/sum[src]) * V[dst]; 16 lanes/edge
__global__ void edge_aggr(const float* __restrict__ V, const float* __restrict__ sc,
                          const float* __restrict__ sum, const int* __restrict__ src,
                          const int* __restrict__ dst, float* __restrict__ out,
                          int colbase, int E) {
  int t = blockIdx.x * 256 + threadIdx.x;
  int e = t >> 4, sub = t & 15;
  if (e >= E) return;
  int s = src[e], d = dst[e];
  float a = sc[e] / sum[s];
  float4 v = *(const float4*)(V + (size_t)d * 64 + sub * 4);
  float* o = out + (size_t)s * 128 + colbase + sub * 4;
  atomicAdd(o + 0, a * v.x);
  atomicAdd(o + 1, a * v.y);
  atomicAdd(o + 2, a * v.z);
  atomicAdd(o + 3, a * v.w);
}

__global__ void add_ew(const float* __restrict__ a, const float* __restrict__ b,
                       float* __restrict__ o, int n) {
  int i = blockIdx.x * 256 + threadIdx.x;
  if (i < n) o[i] = a[i] + b[i];
}

__global__ void fill_u32(unsigned* __restrict__ p, unsigned v, int n) {
  int i = blockIdx.x * 256 + threadIdx.x;
  if (i < n) p[i] = v;
}

// softmax over 16 classes, 16 lanes per row
__global__ void softmax16(const float* __restrict__ logits, float* __restrict__ out,
                          int rows) {
  int t = blockIdx.x * 256 + threadIdx.x;
  int r = t >> 4, c = t & 15;
  if (r >= rows) return;
  float v = logits[(size_t)r * 16 + c];
  float m = v;
  for (int o = 1; o < 16; o <<= 1) m = fmaxf(m, __shfl_xor(m, o));
  float e = expf(v - m);
  float s = e;
  for (int o = 1; o < 16; o <<= 1) s += __shfl_xor(s, o);
  out[(size_t)r * 16 + c] = e / s;
}

// ---------------------------------------------------------------------------
extern "C" void kernel_launch(void* const* d_in, const int* in_sizes, int n_in,
                              void* d_out, int out_size, void* d_ws, size_t ws_size,
                              hipStream_t stream) {
  const float* x        = (const float*)d_in[0];
  const float* Q        = (const float*)d_in[1];
  const float* prelin_w = (const float*)d_in[2];
  const float* prelin_b = (const float*)d_in[3];
  const float* bn0_g    = (const float*)d_in[4];
  const float* bn0_b    = (const float*)d_in[5];
  const float* bn0_m    = (const float*)d_in[6];
  const float* bn0_v    = (const float*)d_in[7];
  const float* mmpn_w   = (const float*)d_in[8];
  const float* mmpn_b   = (const float*)d_in[9];
  const float* mbn_g    = (const float*)d_in[10];
  const float* mbn_b    = (const float*)d_in[11];
  const float* mbn_m    = (const float*)d_in[12];
  const float* mbn_v    = (const float*)d_in[13];
  const float* A_val    = (const float*)d_in[14];
  const float* imp_val  = (const float*)d_in[15];
  const float* rowv_w   = (const float*)d_in[16];
  const float* rowv_b   = (const float*)d_in[17];
  const float* colv_w   = (const float*)d_in[18];
  const float* colv_b   = (const float*)d_in[19];
  const float* rowq_w   = (const float*)d_in[20];
  const float* rowq_b   = (const float*)d_in[21];
  const float* colq_w   = (const float*)d_in[22];
  const float* colq_b   = (const float*)d_in[23];
  const float* pbn_g    = (const float*)d_in[24];
  const float* pbn_b    = (const float*)d_in[25];
  const float* pbn_m    = (const float*)d_in[26];
  const float* pbn_v    = (const float*)d_in[27];
  const float* final_w  = (const float*)d_in[28];
  const float* final_b  = (const float*)d_in[29];
  const int* A_src   = (const int*)d_in[30];
  const int* A_dst   = (const int*)d_in[31];
  const int* imp_src = (const int*)d_in[32];
  const int* imp_dst = (const int*)d_in[33];
  const int* row_src = (const int*)d_in[34];
  const int* row_dst = (const int*)d_in[35];
  const int* col_src = (const int*)d_in[36];
  const int* col_dst = (const int*)d_in[37];
  float* out = (float*)d_out;

  // ---- workspace carve (floats) ----
  float* W = (float*)d_ws;
  size_t o = 0;
  float* B0   = W + o; o += (size_t)N_PIX * HIDE_;   // x0, later hyp
  float* B1   = W + o; o += (size_t)N_PIX * HIDE_;   // px
  float* B2   = W + o; o += (size_t)N_PIX * HIDE_;   // pixel-layer accumulator / H1
  float* rowv = W + o; o += (size_t)N_PIX * OUT_;
  float* colv = W + o; o += (size_t)N_PIX * OUT_;
  float* rowQ = W + o; o += (size_t)N_PIX * OUT_;
  float* colQ = W + o; o += (size_t)N_PIX * OUT_;
  float* rsc  = W + o; o += (size_t)E_PX_;           // row scores, later logits
  float* csc  = W + o; o += (size_t)E_PX_;           // col scores
  unsigned* rmaxu = (unsigned*)(W + o); o += N_PIX;
  unsigned* cmaxu = (unsigned*)(W + o); o += N_PIX;
  float* rsum = W + o; o += N_PIX;
  float* csum = W + o; o += N_PIX;
  float* Qs   = W + o; o += S_SP;
  float* QtX  = W + o; o += (size_t)S_SP * HIDE_;    // later mmpn GEMM scratch h
  float* hpA  = W + o; o += (size_t)S_SP * HIDE_;
  float* hpB  = W + o; o += (size_t)S_SP * HIDE_;
  float* agg  = W + o; o += (size_t)S_SP * HIDE_;
  (void)ws_size; (void)in_sizes; (void)n_in; (void)out_size;

  // ---- 1. x0 = BN(x @ prelin_w + b); also seed px ----
  gemm_wmma<<<dim3(N_PIX / 16), dim3(256), 0, stream>>>(
      x, prelin_w, prelin_b, B0, 128, 128, 1, bn0_g, bn0_b, bn0_m, bn0_v, B1);

  // ---- 2. hp = (Q / colsum(Q))^T @ x0 ----
  hipMemsetAsync(Qs, 0, (S_SP + (size_t)S_SP * HIDE_) * sizeof(float), stream);  // Qs + QtX
  colsum_q<<<dim3(S_SP / 256, 64), dim3(256), 0, stream>>>(Q, Qs);
  qtx_wmma<<<dim3(S_SP / 16, 64), dim3(256), 0, stream>>>(Q, B0, QtX, N_PIX / 64);
  scale_hp<<<dim3(S_SP * HIDE_ / 256), dim3(256), 0, stream>>>(QtX, Qs, hpA);

  // ---- 3. 5 MMPN layers ----
  float* hp_cur = hpA;
  float* hp_nxt = hpB;
  for (int i = 0; i < 5; ++i) {
    gemm_wmma<<<dim3(S_SP / 16), dim3(256), 0, stream>>>(
        hp_cur, mmpn_w + (size_t)i * HIDE_ * HIDE_, mmpn_b + (size_t)i * HIDE_,
        QtX, 128, 128, 0, nullptr, nullptr, nullptr, nullptr, nullptr);
    hipMemsetAsync(agg, 0, (size_t)S_SP * HIDE_ * sizeof(float), stream);
    edge_scatter128<<<dim3(E_A_ * 32 / 256), dim3(256), 0, stream>>>(
        QtX, A_val, A_src, A_dst, E_A_, agg);
    edge_scatter128<<<dim3(E_IMP_ * 32 / 256), dim3(256), 0, stream>>>(
        hp_cur, imp_val, imp_src, imp_dst, E_IMP_, agg);
    bn_lrelu128<<<dim3(S_SP * HIDE_ / 256), dim3(256), 0, stream>>>(
        agg, hp_nxt, mbn_g + i * HIDE_, mbn_b + i * HIDE_, mbn_m + i * HIDE_,
        mbn_v + i * HIDE_, S_SP * HIDE_);
    float* t = hp_cur; hp_cur = hp_nxt; hp_nxt = t;
  }

  // ---- 4. hyp = Q @ hp  (reuse B0; x0 no longer needed) ----
  gemm_wmma<<<dim3(N_PIX / 16), dim3(256), 0, stream>>>(
      Q, hp_cur, nullptr, B0, S_SP, 128, 0, nullptr, nullptr, nullptr, nullptr,
      nullptr);

  // ---- 5. pixel sparse-former, 2 layers ----
  for (int j = 0; j < 2; ++j) {
    const size_t wj = (size_t)j * HIDE_ * OUT_;
    const size_t bj = (size_t)j * OUT_;
    gemm_wmma<<<dim3(N_PIX / 16), dim3(128), 0, stream>>>(
        B1, rowv_w + wj, rowv_b + bj, rowv, 128, 64, 0, nullptr, nullptr, nullptr,
        nullptr, nullptr);
    gemm_wmma<<<dim3(N_PIX / 16), dim3(128), 0, stream>>>(
        B1, colv_w + wj, colv_b + bj, colv, 128, 64, 0, nullptr, nullptr, nullptr,
        nullptr, nullptr);
    gemm_wmma<<<dim3(N_PIX / 16), dim3(128), 0, stream>>>(
        B1, rowq_w + wj, rowq_b + bj, rowQ, 128, 64, 0, nullptr, nullptr, nullptr,
        nullptr, nullptr);
    gemm_wmma<<<dim3(N_PIX / 16), dim3(128), 0, stream>>>(
        B1, colq_w + wj, colq_b + bj, colQ, 128, 64, 0, nullptr, nullptr, nullptr,
        nullptr, nullptr);
    ln64<<<dim3(N_PIX * 32 / 256), dim3(256), 0, stream>>>(rowQ, N_PIX);
    ln64<<<dim3(N_PIX * 32 / 256), dim3(256), 0, stream>>>(colQ, N_PIX);

    fill_u32<<<dim3(2 * N_PIX / 256), dim3(256), 0, stream>>>(
        rmaxu, 0x007FFFFFu /* fmono(-inf) */, 2 * N_PIX);          // rmaxu+cmaxu
    hipMemsetAsync(rsum, 0, 2 * (size_t)N_PIX * sizeof(float), stream);  // rsum+csum
    hipMemsetAsync(B2, 0, (size_t)N_PIX * HIDE_ * sizeof(float), stream);

    edge_scores<<<dim3(E_PX_ * 16 / 256), dim3(256), 0, stream>>>(
        rowQ, row_src, row_dst, rsc, rmaxu, E_PX_);
    edge_scores<<<dim3(E_PX_ * 16 / 256), dim3(256), 0, stream>>>(
        colQ, col_src, col_dst, csc, cmaxu, E_PX_);
    edge_expsum<<<dim3(E_PX_ / 256), dim3(256), 0, stream>>>(
        rsc, row_src, rmaxu, rsum, E_PX_);
    edge_expsum<<<dim3(E_PX_ / 256), dim3(256), 0, stream>>>(
        csc, col_src, cmaxu, csum, E_PX_);
    edge_aggr<<<dim3(E_PX_ * 16 / 256), dim3(256), 0, stream>>>(
        rowv, rsc, rsum, row_src, row_dst, B2, 0, E_PX_);
    edge_aggr<<<dim3(E_PX_ * 16 / 256), dim3(256), 0, stream>>>(
        colv, csc, csum, col_src, col_dst, B2, 64, E_PX_);

    bn_lrelu128<<<dim3(N_PIX * HIDE_ / 256), dim3(256), 0, stream>>>(
        B2, B1, pbn_g + j * HIDE_, pbn_b + j * HIDE_, pbn_m + j * HIDE_,
        pbn_v + j * HIDE_, N_PIX * HIDE_);
  }

  // ---- 6. H1 = hyp + px; softmax(H1 @ final_w + b) ----
  add_ew<<<dim3(N_PIX * HIDE_ / 256), dim3(256), 0, stream>>>(
      B0, B1, B2, N_PIX * HIDE_);
  gemm_wmma<<<dim3(N_PIX / 16), dim3(32), 0, stream>>>(
      B2, final_w, final_b, rsc /* logits (reuse) */, 128, 16, 0, nullptr, nullptr,
      nullptr, nullptr, nullptr);
  softmax16<<<dim3(N_PIX * 16 / 256), dim3(256), 0, stream>>>(rsc, out, N_PIX);
}